// BernoulliSampling_53953379172958
// MI455X (gfx1250) — compile-verified
//
#include <hip/hip_runtime.h>
#include <stdint.h>

// BernoulliSampling: out = mean_s( u_s < sigmoid(z) ), u = threefry2x32 uniforms
// (exact jax.random.uniform(key=42) bit layout), z: 8192x4096 f32, S=10.
//
// Memory floor: 268 MB @ 23.3 TB/s ~= 11.5 us. Exact-RNG compute (~360 int
// VALU ops/elem) dominates -> compute-bound. Data path: CDNA5 async
// global->LDS DMA double-buffering (ASYNCcnt) + L2 prefetch; S==10 case is
// device-dispatched to a fully unrolled template so the 5 threefry calls per
// element schedule as one straight-line VALU block. No WMMA (elementwise op).

#define TPB 256
#define NBLOCKS 8192

typedef int v4i __attribute__((ext_vector_type(4)));

#if defined(__has_builtin)
#if __has_builtin(__builtin_amdgcn_global_load_async_to_lds_b128)
#define HAVE_ASYNC_B128 1
#endif
#if __has_builtin(__builtin_amdgcn_s_wait_asynccnt)
#define HAVE_WAIT_ASYNC 1
#endif
#endif

// ---- CDNA5 async global->LDS (ASYNCcnt-tracked) -------------------------
__device__ __forceinline__ void cdna5_async_load_b128(uint64_t gbase,
                                                      uint32_t byte_off,
                                                      uint32_t lds_off) {
#if defined(HAVE_ASYNC_B128)
  __builtin_amdgcn_global_load_async_to_lds_b128(
      (__attribute__((address_space(1))) v4i*)(gbase + (uint64_t)byte_off),
      (__attribute__((address_space(3))) v4i*)lds_off,
      /*offset=*/0, /*cpol=*/0);
#else
  asm volatile("global_load_async_to_lds_b128 %0, %1, %2"
               :: "v"(lds_off), "v"(byte_off), "s"(gbase)
               : "memory");
#endif
}

__device__ __forceinline__ void cdna5_wait_async0() {
#if defined(HAVE_WAIT_ASYNC)
  __builtin_amdgcn_s_wait_asynccnt(0);
#else
  asm volatile("s_wait_asynccnt 0" ::: "memory");
#endif
}
__device__ __forceinline__ void cdna5_wait_async1() {
#if defined(HAVE_WAIT_ASYNC)
  __builtin_amdgcn_s_wait_asynccnt(1);
#else
  asm volatile("s_wait_asynccnt 1" ::: "memory");
#endif
}
__device__ __forceinline__ void cdna5_wait_ds0() {
  asm volatile("s_wait_dscnt 0" ::: "memory");
}

// ---- threefry2x32-20, key = jax.random.key(42) => (k0,k1) = (0,42) ------
__device__ __forceinline__ void threefry2x32_seed42(uint32_t x0, uint32_t x1,
                                                    uint32_t& o0, uint32_t& o1) {
  const uint32_t ks0 = 0u;
  const uint32_t ks1 = 42u;
  const uint32_t ks2 = 0x1BD11BDAu ^ ks0 ^ ks1;
  x0 += ks0; x1 += ks1;
#define TF_R(r) { x0 += x1; x1 = __builtin_rotateleft32(x1, r); x1 ^= x0; }
  TF_R(13) TF_R(15) TF_R(26) TF_R(6)
  x0 += ks1; x1 += ks2 + 1u;
  TF_R(17) TF_R(29) TF_R(16) TF_R(24)
  x0 += ks2; x1 += ks0 + 2u;
  TF_R(13) TF_R(15) TF_R(26) TF_R(6)
  x0 += ks0; x1 += ks1 + 3u;
  TF_R(17) TF_R(29) TF_R(16) TF_R(24)
  x0 += ks1; x1 += ks2 + 4u;
  TF_R(13) TF_R(15) TF_R(26) TF_R(6)
  x0 += ks2; x1 += ks0 + 5u;
#undef TF_R
  o0 = x0; o1 = x1;
}

// jax uniform: ((bits >> 9) | 0x3f800000).view(f32) - 1.0  -> [0,1)
__device__ __forceinline__ float u01(uint32_t bits) {
  return __uint_as_float((bits >> 9) | 0x3f800000u) - 1.0f;
}

// Element j (flattened over [B,D]); sample s<halfS uses x0 of ctr
// (s*M+j, (s+halfS)*M+j); sample s+halfS uses x1 of the same call.
// HS>0: compile-time halfS (fully unrolled). HS==0: runtime halfS.
template <uint32_t HS>
__device__ __forceinline__ float bern_mean(float zv, uint32_t j, uint32_t M,
                                           uint32_t halfS, float fS) {
  const float p = 1.0f / (1.0f + __expf(-zv));  // sigmoid
  uint32_t cnt = 0u;
  if constexpr (HS > 0) {
    uint32_t c1base = j + HS * M;
#pragma unroll
    for (uint32_t s = 0; s < HS; ++s) {
      uint32_t r0, r1;
      threefry2x32_seed42(j + s * M, c1base + s * M, r0, r1);
      cnt += (uint32_t)(u01(r0) < p);
      cnt += (uint32_t)(u01(r1) < p);
    }
  } else {
    uint32_t c0 = j;
    uint32_t c1 = j + halfS * M;
    for (uint32_t s = 0; s < halfS; ++s) {
      uint32_t r0, r1;
      threefry2x32_seed42(c0, c1, r0, r1);
      cnt += (uint32_t)(u01(r0) < p);
      cnt += (uint32_t)(u01(r1) < p);
      c0 += M; c1 += M;
    }
  }
  return (float)cnt / fS;
}

template <uint32_t HS>
__device__ __forceinline__ void run_pipeline(float4 (&tile)[2][TPB],
                                             const float* __restrict__ z,
                                             float* __restrict__ out,
                                             uint32_t M, uint32_t n4,
                                             uint32_t v, uint32_t stride,
                                             uint32_t my0, uint32_t my1,
                                             uint32_t tid,
                                             uint32_t halfS, float fS) {
  const uint64_t gbase = (uint64_t)(uintptr_t)z;

  // Prologue: start stage-0 DMA for this thread's first tile.
  if (v < n4) cdna5_async_load_b128(gbase, v << 4, my0);

  uint32_t stage = 0;
  for (; v < n4; v += stride, stage ^= 1) {
    const uint32_t vnext = v + stride;
    if (vnext < n4) {
      // Next tile into the other buffer, keep it in flight (<=1 wait),
      // and warm L2 two strides ahead.
      cdna5_async_load_b128(gbase, vnext << 4, stage ? my0 : my1);
      if (vnext + stride < n4)
        __builtin_prefetch(&z[(size_t)(vnext + stride) << 2], 0, 1);
      cdna5_wait_async1();   // oldest (current) DMA complete; next in flight
    } else {
      cdna5_wait_async0();   // drain: current tile complete
    }

    const float4 zv = tile[stage][tid];  // ds_read_b128 from my own slot
    cdna5_wait_ds0();                    // regs hold data before buffer recycle

    float4 o;
    const uint32_t j0 = v << 2;          // flat element index of zv.x
    o.x = bern_mean<HS>(zv.x, j0 + 0u, M, halfS, fS);
    o.y = bern_mean<HS>(zv.y, j0 + 1u, M, halfS, fS);
    o.z = bern_mean<HS>(zv.z, j0 + 2u, M, halfS, fS);
    o.w = bern_mean<HS>(zv.w, j0 + 3u, M, halfS, fS);

    reinterpret_cast<float4*>(out)[v] = o;  // global_store_b128
  }
}

__global__ void __launch_bounds__(TPB)
BernoulliSampling_53953379172958_kernel(const float* __restrict__ z,
                                        float* __restrict__ out,
                                        const int* __restrict__ nsamp,
                                        uint32_t M) {
  __shared__ float4 tile[2][TPB];  // 8 KB double buffer, 16 B/lane/stage

  const uint32_t n4     = M >> 2;
  const uint32_t tid    = threadIdx.x;
  const uint32_t stride = gridDim.x * TPB;        // in float4 units
  const uint32_t v      = blockIdx.x * TPB + tid;

  const int      S     = *nsamp;                  // = 10 (device-read scalar)
  const uint32_t halfS = (uint32_t)S >> 1;
  const float    fS    = (float)S;

  const uint32_t lds_base = (uint32_t)(uintptr_t)&tile[0][0]; // low 32b = LDS off
  const uint32_t my0      = lds_base + tid * 16u;             // my slot, stage 0
  const uint32_t my1      = my0 + TPB * 16u;                  // my slot, stage 1

  // Uniform scalar dispatch: S==10 gets the fully unrolled specialization.
  if (halfS == 5u)
    run_pipeline<5>(tile, z, out, M, n4, v, stride, my0, my1, tid, halfS, fS);
  else
    run_pipeline<0>(tile, z, out, M, n4, v, stride, my0, my1, tid, halfS, fS);
}

extern "C" void kernel_launch(void* const* d_in, const int* in_sizes, int n_in,
                              void* d_out, int out_size, void* d_ws, size_t ws_size,
                              hipStream_t stream) {
  const float* z     = (const float*)d_in[0];
  const int*   nsamp = (const int*)d_in[1];   // scalar num_samples on device
  float*       out   = (float*)d_out;
  const uint32_t M   = (uint32_t)in_sizes[0]; // 8192*4096 = 33,554,432

  dim3 grid(NBLOCKS), block(TPB);
  BernoulliSampling_53953379172958_kernel<<<grid, block, 0, stream>>>(z, out, nsamp, M);
  (void)d_ws; (void)ws_size; (void)n_in; (void)out_size;
}